// CoordinalizationCell_28372553957615
// MI455X (gfx1250) — compile-verified
//
#include <hip/hip_runtime.h>
#include <stdint.h>

// ---------------------------------------------------------------------------
// Backbone NeRF scan: B=16384 chains x T=512 steps, 1 thread per chain.
// Memory-bound (448 MB @ 23.3 TB/s ~= 19 us); compute hides under traffic.
// CDNA5 specifics: async global->LDS staging of the angle tiles (ASYNCcnt),
// double-buffered, wave32 blocks, b128 stores.
// ---------------------------------------------------------------------------

constexpr int BLOCK   = 128;                 // 4 wave32s per block
constexpr int T_STEPS = 512;
constexpr int TCHUNK  = 8;                   // steps staged per tile
constexpr int NCHUNK  = T_STEPS / TCHUNK;    // 64
constexpr int ROW_F   = 18;                  // 16 payload floats + 2 pad (bank-conflict-free)

// Geometry constants, pre-folded to f32 (matches reference np.float32 math):
#define BOND_CN_R   (1.0f / 1.32f)
#define BOND_CAC_R  (1.0f / 1.52f)
#define BOND_NCA_R  (1.0f / 1.45f)
#define D_C_X   (-0.50738649f)   // BOND_CAC * cos(109.5deg)
#define K_C     ( 1.43281508f)   // BOND_CAC * sin(109.5deg)
#define D_O_X   (-0.63349686f)   // BOND_CO  * cos(121.0deg)
#define K_O     ( 1.05431576f)   // BOND_CO  * sin(121.0deg)
#define D_N_X   (-0.57864985f)   // BOND_CN  * cos(116.0deg)
#define K_N     ( 1.18640808f)   // BOND_CN  * sin(116.0deg)
#define D_A_X   (-0.76838300f)   // BOND_NCA * cos(122.0deg)
#define D_A_Y   (-1.22966975f)   // BOND_NCA * sin(122.0deg) * cos(-fl32(pi))
#define D_A_Z   ( 1.0750e-7f)    // BOND_NCA * sin(122.0deg) * sin(-fl32(pi))

#if defined(__HIP_DEVICE_COMPILE__) && \
    __has_builtin(__builtin_amdgcn_global_load_async_to_lds_b128) && \
    __has_builtin(__builtin_amdgcn_s_wait_asynccnt)
typedef int v4i __attribute__((vector_size(16)));
typedef __attribute__((address_space(1))) v4i g_v4i;   // global b128 payload
typedef __attribute__((address_space(3))) v4i l_v4i;   // LDS b128 payload
#define ASYNC_CP16(gsrc, ldst)                                           \
  __builtin_amdgcn_global_load_async_to_lds_b128(                        \
      (g_v4i*)(uintptr_t)(gsrc), (l_v4i*)(uintptr_t)(ldst), 0, 0)
#define WAIT_ASYNC(n) __builtin_amdgcn_s_wait_asynccnt(n)
#else
// Fallback: synchronous copy through VGPRs (global_load_b128 + ds_store_b128).
#define ASYNC_CP16(gsrc, ldst) do { *(float4*)(ldst) = *(const float4*)(gsrc); } while (0)
#define WAIT_ASYNC(n) do { } while (0)
#endif

struct F3 { float x, y, z; };

__device__ __forceinline__ F3 sub3(F3 a, F3 b) { return {a.x - b.x, a.y - b.y, a.z - b.z}; }
__device__ __forceinline__ F3 scl3(F3 a, float s) { return {a.x * s, a.y * s, a.z * s}; }
__device__ __forceinline__ F3 crs3(F3 a, F3 b) {
  return {a.y * b.z - a.z * b.y, a.z * b.x - a.x * b.z, a.x * b.y - a.y * b.x};
}
__device__ __forceinline__ F3 nrm3(F3 v) {
  float inv = rsqrtf(v.x * v.x + v.y * v.y + v.z * v.z);   // v_rsq_f32 path
  return scl3(v, inv);
}
// p + a*sa + b*sb + c*sc, fully fused
__device__ __forceinline__ F3 comb3(F3 p, F3 a, float sa, F3 b, float sb, F3 c, float sc) {
  return { fmaf(a.x, sa, fmaf(b.x, sb, fmaf(c.x, sc, p.x))),
           fmaf(a.y, sa, fmaf(b.y, sb, fmaf(c.y, sc, p.y))),
           fmaf(a.z, sa, fmaf(b.z, sb, fmaf(c.z, sc, p.z))) };
}

__global__ __launch_bounds__(BLOCK)
void backbone_scan_kernel(const float* __restrict__ x,
                          const float* __restrict__ sC,
                          const float* __restrict__ sN,
                          const float* __restrict__ sCa,
                          float* __restrict__ out)
{
  // Double-buffered angle tile: [buf][row(=chain) * ROW_F + 2*t_local]
  __shared__ float xt[2][BLOCK * ROW_F];

  const int tid = threadIdx.x;
  const int b0  = blockIdx.x * BLOCK;
  const int b   = b0 + tid;

  const float* gx = x + (size_t)b0 * (T_STEPS * 2);

  F3 C  = { sC [3 * b + 0], sC [3 * b + 1], sC [3 * b + 2] };
  F3 N  = { sN [3 * b + 0], sN [3 * b + 1], sN [3 * b + 2] };
  F3 Ca = { sCa[3 * b + 0], sCa[3 * b + 1], sCa[3 * b + 2] };

  float* outp = out + (size_t)b * (T_STEPS * 12);

  // Prefetch chunk 0: 128 rows x 64 B = 512 x 16 B segments, 4 per thread.
#pragma unroll
  for (int k = 0; k < 4; ++k) {
    int idx = tid + k * BLOCK;
    int row = idx >> 2, seg = idx & 3;
    ASYNC_CP16(gx + (size_t)row * (T_STEPS * 2) + seg * 4,
               &xt[0][row * ROW_F + seg * 4]);
  }

  for (int c = 0; c < NCHUNK; ++c) {
    const int buf = c & 1;
    if (c + 1 < NCHUNK) {
      // Issue prefetch of the next tile, then wait for the *previous* 4
      // in-order async loads (current tile) to land in LDS.
#pragma unroll
      for (int k = 0; k < 4; ++k) {
        int idx = tid + k * BLOCK;
        int row = idx >> 2, seg = idx & 3;
        ASYNC_CP16(gx + (size_t)row * (T_STEPS * 2) + (size_t)(c + 1) * (TCHUNK * 2) + seg * 4,
                   &xt[buf ^ 1][row * ROW_F + seg * 4]);
      }
      WAIT_ASYNC(4);
    } else {
      WAIT_ASYNC(0);
    }
    __syncthreads();   // everyone's async loads for tile c are now visible

    const float* xr = &xt[buf][tid * ROW_F];
#pragma unroll 2
    for (int tt = 0; tt < TCHUNK; ++tt) {
      float x0 = xr[2 * tt + 0];
      float x1 = xr[2 * tt + 1];
      float s0 = __sinf(x0), c0 = __cosf(x0);   // v_sin/v_cos (|x| <= pi)
      float s1 = __sinf(x1), c1 = __cosf(x1);

      // ---- place C ----
      F3 bc = scl3(sub3(Ca, N), BOND_NCA_R);
      F3 nk = nrm3(crs3(sub3(N, C), bc));
      F3 m1 = crs3(nk, bc);
      F3 Cc = comb3(Ca, bc, D_C_X, m1, K_C * c0, nk, K_C * s0);

      // ---- O and next N share the same frame (bc, nk, m1) ----
      bc = scl3(sub3(Cc, Ca), BOND_CAC_R);
      nk = nrm3(crs3(sub3(Ca, N), bc));
      m1 = crs3(nk, bc);
      // D_O uses cos/sin(x1+pi) = (-c1, -s1)
      F3 Co = comb3(Cc, bc, D_O_X, m1, -K_O * c1, nk, -K_O * s1);
      F3 Cn = comb3(Cc, bc, D_N_X, m1,  K_N * c1, nk,  K_N * s1);

      // ---- place next Ca ----
      bc = scl3(sub3(Cn, Cc), BOND_CN_R);
      nk = nrm3(crs3(sub3(Cc, Ca), bc));
      m1 = crs3(nk, bc);
      F3 Ca2 = comb3(Cn, bc, D_A_X, m1, D_A_Y, nk, D_A_Z);

      // ---- emit 12 floats as 3x b128 stores (lane-contiguous 48 B) ----
      float4* o = (float4*)(outp + (size_t)(c * TCHUNK + tt) * 12);
      o[0] = make_float4(Cc.x, Cc.y, Cc.z, Co.x);
      o[1] = make_float4(Co.y, Co.z, Cn.x, Cn.y);
      o[2] = make_float4(Cn.z, Ca2.x, Ca2.y, Ca2.z);

      C = Cc; N = Cn; Ca = Ca2;
    }
    __syncthreads();   // protect buf from being overwritten by next prefetch
  }
}

extern "C" void kernel_launch(void* const* d_in, const int* in_sizes, int n_in,
                              void* d_out, int out_size, void* d_ws, size_t ws_size,
                              hipStream_t stream) {
  const float* x   = (const float*)d_in[0];   // (B, T, 2) f32
  const float* sC  = (const float*)d_in[1];   // (B, 3)    f32
  const float* sN  = (const float*)d_in[2];   // (B, 3)    f32
  const float* sCa = (const float*)d_in[3];   // (B, 3)    f32
  float* out = (float*)d_out;                 // (B, T, 12) f32

  int B = in_sizes[1] / 3;                    // 16384
  int grid = (B + BLOCK - 1) / BLOCK;         // 128 blocks x 4 waves
  backbone_scan_kernel<<<dim3(grid), dim3(BLOCK), 0, stream>>>(x, sC, sN, sCa, out);
}